// AConvCircular3D_25391846654374
// MI455X (gfx1250) — compile-verified
//
#include <hip/hip_runtime.h>

typedef float v2f __attribute__((ext_vector_type(2)));
typedef float v8f __attribute__((ext_vector_type(8)));
typedef unsigned int u32x4 __attribute__((ext_vector_type(4)));
typedef unsigned int u32x8 __attribute__((ext_vector_type(8)));

// D = A(16x4) * B(4x16) + C, full fp32 (ISA V_WMMA_F32_16X16X4_F32)
#define WMMA4(a, b, c) \
    __builtin_amdgcn_wmma_f32_16x16x4_f32(false, (a), false, (b), (short)0, (c), false, false)

#define NPOS   4096   // 16*16*16
#define CIN    32
#define KDIM   864    // 32*27

// ---------------------------------------------------------------------------
// Kernel 1: fused implicit-GEMM circular conv (init 32ch + qkv 96ch)
// grid = 2*16*16 (b,d,h rows), block = 256 (8 waves, one 16-oc M-tile each)
// x slab staged into LDS by the Tensor Data Mover (9 2D tiles, padded rows)
// ---------------------------------------------------------------------------
__global__ __launch_bounds__(256)
void conv_qkv_kernel(const float* __restrict__ x,
                     const float* __restrict__ w_init,
                     const float* __restrict__ w_qkv,
                     float* __restrict__ out,     // (2,64,4096), channels 0..31
                     float* __restrict__ ws_q,    // (2,32,4096) pre-scaled
                     float* __restrict__ ws_k,    // (2,32,4096)
                     float* __restrict__ ws_v)    // (2,32,4096)
{
    __shared__ float xs[CIN * 3 * 3 * 16];       // [ic][kd][kh][w]

    const int tid  = threadIdx.x;
    const int lane = tid & 31;
    const int wv   = tid >> 5;                   // M-tile 0..7
    const int col  = lane & 15;
    const int hi   = lane >> 4;                  // half-wave
    const int sel  = hi << 1;                    // K-slot offset 0 / 2

    const int blk = blockIdx.x;                  // 0..511
    const int b   = blk >> 8;
    const int d   = (blk >> 4) & 15;
    const int h   = blk & 15;

    // --- TDM staging: for each (kd,kh), DMA a 2D tile of 32 ic-rows x 16 w
    //     floats (row stride 4096 floats) into xs with a 128-DWORD LDS pad
    //     after every 16 DWORDs, interleaving into xs[ic][kd][kh][w].
    if (wv == 0) {
        // D# group 1: data_size=4B(2), pad_enable, pad_interval=16dw(3),
        // pad_amount=128dw(127); tensor 16 x 32, tile 16 x 32, stride 4096.
        u32x8 g1;
        g1[0] = (2u << 16) | (1u << 20) | (3u << 22) | (127u << 25);
        g1[1] = (16u << 16);       // tensor_dim0 = 16  (bits 79:48)
        g1[2] = (32u << 16);       // tensor_dim1 = 32  (bits 111:80)
        g1[3] = (16u << 16);       // tile_dim0   = 16  (bits 127:112)
        g1[4] = 32u;               // tile_dim1   = 32  (bits 143:128)
        g1[5] = 4096u;             // tensor_dim0_stride = 4096 (bits 207:160)
        g1[6] = 0u;
        g1[7] = 0u;
        for (int kd = 0; kd < 3; ++kd) {
            for (int kh = 0; kh < 3; ++kh) {
                const int dz = (d + kd + 15) & 15;
                const int hy = (h + kh + 15) & 15;
                const unsigned long long gaddr =
                    (unsigned long long)(const void*)
                    (x + ((((unsigned long long)b * CIN) * 16 + dz) * 16 + hy) * 16);
                const unsigned lds_byte =
                    (unsigned)(unsigned long long)(const void*)xs + (kd * 3 + kh) * 64u;
                u32x4 g0;
                g0[0] = 1u;                                   // count=1, user D#
                g0[1] = lds_byte;                             // lds_addr
                g0[2] = (unsigned)(gaddr & 0xFFFFFFFFull);    // global_addr lo
                g0[3] = (unsigned)((gaddr >> 32) & 0x01FFFFFFull) | 0x80000000u; // type=2
                asm volatile("tensor_load_to_lds %0, %1" :: "s"(g0), "s"(g1) : "memory");
            }
        }
        __builtin_amdgcn_s_wait_tensorcnt(0);
    }
    __syncthreads();

    // A rows: lane&15 selects output channel within this wave's tile
    const float* wbase = (wv < 2) ? w_init : w_qkv;
    const int    trow  = ((wv < 2) ? wv : (wv - 2)) * 16 + col;
    const float* wrow  = wbase + trow * KDIM;    // (oc, ic*27 + tap)

    v8f acc = {};
    for (int tap = 0; tap < 27; ++tap) {
        const int kd = tap / 9;
        const int kh = (tap / 3) % 3;
        const int kw = tap % 3;
        const int wn = (col + kw + 15) & 15;     // circular w for B column
        const int xbase = (kd * 3 + kh) * 16 + wn;
#pragma unroll
        for (int ic0 = 0; ic0 < CIN; ic0 += 4) {
            v2f a, bb;
            a.x  = wrow[(ic0 + sel)     * 27 + tap];
            a.y  = wrow[(ic0 + sel + 1) * 27 + tap];
            bb.x = xs[(ic0 + sel)     * 144 + xbase];
            bb.y = xs[(ic0 + sel + 1) * 144 + xbase];
            acc  = WMMA4(a, bb, acc);
        }
    }

    const int p = (d * 16 + h) * 16 + col;       // spatial position of this column
#pragma unroll
    for (int i = 0; i < 8; ++i) {
        const int goc = wv * 16 + i + hi * 8;    // 0..127
        const float val = acc[i];
        if (goc < 32) {
            out[(b * 64 + goc) * NPOS + p] = val;
        } else {
            const int qc = goc - 32;
            if (qc < 32)       ws_q[(b * 32 + qc)        * NPOS + p] = val * 0.35355339059327373f;
            else if (qc < 64)  ws_k[(b * 32 + (qc - 32)) * NPOS + p] = val;
            else               ws_v[(b * 32 + (qc - 64)) * NPOS + p] = val;
        }
    }
}

// ---------------------------------------------------------------------------
// Kernel 2: flash attention, no materialized N x N logits.
// grid = 2*4*32 (b, head, 128-row chunk), block = 256 (8 waves, 16 rows each)
// ---------------------------------------------------------------------------
__global__ __launch_bounds__(256)
void flash_attn_kernel(const float* __restrict__ qg,
                       const float* __restrict__ kg,
                       const float* __restrict__ vg,
                       float* __restrict__ attn_out)  // flat (b,H,N,dv_h)
{
    __shared__ float lds_p[8][16][16];           // per-wave P^T bounce tile

    const int tid  = threadIdx.x;
    const int lane = tid & 31;
    const int wv   = tid >> 5;
    const int col  = lane & 15;
    const int hi   = lane >> 4;
    const int sel  = hi << 1;

    const int blk   = blockIdx.x;                // 0..255
    const int b     = blk >> 7;
    const int head  = (blk >> 5) & 3;
    const int chunk = blk & 31;
    const int n0    = chunk * 128 + wv * 16;

    const float* qh = qg + (b * 4 + head) * 8 * NPOS;  // [c][n], already scaled
    const float* kh = kg + (b * 4 + head) * 8 * NPOS;
    const float* vh = vg + (b * 4 + head) * 8 * NPOS;

    // Q B-fragments for the wave's fixed n-tile (K-steps c0 = 0, 4)
    v2f qb0, qb1;
    qb0.x = qh[(0 + sel) * NPOS + n0 + col];
    qb0.y = qh[(1 + sel) * NPOS + n0 + col];
    qb1.x = qh[(4 + sel) * NPOS + n0 + col];
    qb1.y = qh[(5 + sel) * NPOS + n0 + col];

    float row_max = -3.0e38f;
    float row_sum = 0.0f;
    v8f   o       = {};                          // O accumulator (rows = v)

    for (int m0 = 0; m0 < NPOS; m0 += 16) {
        // pull next m-tile K/V lines toward the WGP while this tile computes
        if (m0 + 16 < NPOS) {
            __builtin_prefetch(kh + (0 + sel) * NPOS + m0 + 16 + col, 0, 0);
            __builtin_prefetch(vh + (col & 7) * NPOS + m0 + 16 + col, 0, 0);
        }

        // S^T tile: rows = m (key), cols = n (query): A = K, B = Q
        v2f ka0, ka1;
        ka0.x = kh[(0 + sel) * NPOS + m0 + col];
        ka0.y = kh[(1 + sel) * NPOS + m0 + col];
        ka1.x = kh[(4 + sel) * NPOS + m0 + col];
        ka1.y = kh[(5 + sel) * NPOS + m0 + col];
        v8f s = {};
        s = WMMA4(ka0, qb0, s);
        s = WMMA4(ka1, qb1, s);

        // online softmax over m (8 rows in-lane + half-wave exchange)
        float tmax = s[0];
#pragma unroll
        for (int i = 1; i < 8; ++i) tmax = fmaxf(tmax, s[i]);
        tmax = fmaxf(tmax, __shfl_xor(tmax, 16, 32));
        const float nm  = fmaxf(row_max, tmax);
        const float fac = __expf(row_max - nm);
        row_max = nm;

        float pr[8];
        float tsum = 0.0f;
#pragma unroll
        for (int i = 0; i < 8; ++i) { pr[i] = __expf(s[i] - nm); tsum += pr[i]; }
        tsum += __shfl_xor(tsum, 16, 32);
        row_sum = row_sum * fac + tsum;
#pragma unroll
        for (int i = 0; i < 8; ++i) o[i] *= fac;

        // bounce P^T through LDS to re-feed as B operand
#pragma unroll
        for (int i = 0; i < 8; ++i) lds_p[wv][i + hi * 8][col] = pr[i];
        asm volatile("s_wait_dscnt 0" ::: "memory");

        // O += V(16x16, rows v>=8 zero) * P^T(16m x 16n); 4 K-steps
#pragma unroll
        for (int k0 = 0; k0 < 16; k0 += 4) {
            v2f av, pb;
            const int vr = col;                  // A row index (both half-waves)
            av.x = (vr < 8) ? vh[vr * NPOS + m0 + k0 + sel]     : 0.0f;
            av.y = (vr < 8) ? vh[vr * NPOS + m0 + k0 + sel + 1] : 0.0f;
            pb.x = lds_p[wv][k0 + sel][col];
            pb.y = lds_p[wv][k0 + sel + 1][col];
            o = WMMA4(av, pb, o);
        }
        asm volatile("s_wait_dscnt 0" ::: "memory");
    }

    // valid output rows v=0..7 live in the lower half-wave
    if (hi == 0) {
        const float inv = 1.0f / row_sum;
        float* dst = attn_out + ((b * 4 + head) * NPOS + (n0 + col)) * 8;
#pragma unroll
        for (int i = 0; i < 8; ++i) dst[i] = o[i] * inv;
    }
}

// ---------------------------------------------------------------------------
// Kernel 3: 1x1x1 projection (32->32) + bias, reading the reinterpret-reshaped
// attention buffer as (b, 32, 4096). grid = 128, block = 256 (8 wave-tasks)
// ---------------------------------------------------------------------------
__global__ __launch_bounds__(256)
void proj_kernel(const float* __restrict__ attn,   // flat (b,32,4096) view
                 const float* __restrict__ w_out,  // (32,32)
                 const float* __restrict__ b_out,  // (32,)
                 float* __restrict__ out)          // (2,64,4096), channels 32..63
{
    const int tid  = threadIdx.x;
    const int lane = tid & 31;
    const int wv   = tid >> 5;
    const int col  = lane & 15;
    const int hi   = lane >> 4;
    const int sel  = hi << 1;

    const int task = blockIdx.x * 8 + wv;        // 0..1023
    const int b    = task >> 9;
    const int rem  = task & 511;
    const int pt   = rem >> 1;                   // 0..255 position tiles
    const int mt   = rem & 1;                    // 0..1 oc tiles
    const int p0   = pt * 16;

    const float* arow = attn + b * (32 * NPOS);
    const float* wrow = w_out + (mt * 16 + col) * 32;

    v8f acc = {};
#pragma unroll
    for (int k0 = 0; k0 < 32; k0 += 4) {
        v2f a, bb;
        a.x  = wrow[k0 + sel];
        a.y  = wrow[k0 + sel + 1];
        bb.x = arow[(k0 + sel)     * NPOS + p0 + col];
        bb.y = arow[(k0 + sel + 1) * NPOS + p0 + col];
        acc  = WMMA4(a, bb, acc);
    }
#pragma unroll
    for (int i = 0; i < 8; ++i) {
        const int oc = mt * 16 + i + hi * 8;
        out[(b * 64 + 32 + oc) * NPOS + p0 + col] = acc[i] + b_out[oc];
    }
}

// ---------------------------------------------------------------------------
extern "C" void kernel_launch(void* const* d_in, const int* in_sizes, int n_in,
                              void* d_out, int out_size, void* d_ws, size_t ws_size,
                              hipStream_t stream)
{
    (void)in_sizes; (void)n_in; (void)out_size; (void)ws_size;
    const float* x      = (const float*)d_in[0];
    const float* w_init = (const float*)d_in[1];
    const float* w_qkv  = (const float*)d_in[2];
    const float* w_out  = (const float*)d_in[3];
    const float* b_out  = (const float*)d_in[4];
    float* out = (float*)d_out;

    float* ws      = (float*)d_ws;               // needs 4 MB
    float* ws_q    = ws;                         // 2*32*4096
    float* ws_k    = ws + 262144;
    float* ws_v    = ws + 524288;
    float* ws_attn = ws + 786432;

    conv_qkv_kernel  <<<512, 256, 0, stream>>>(x, w_init, w_qkv, out, ws_q, ws_k, ws_v);
    flash_attn_kernel<<<256, 256, 0, stream>>>(ws_q, ws_k, ws_v, ws_attn);
    proj_kernel      <<<128, 256, 0, stream>>>(ws_attn, w_out, b_out, out);
}